// HSTU_BSA_Triton_5119601017309
// MI455X (gfx1250) — compile-verified
//
#include <hip/hip_runtime.h>
#include <hip/hip_bf16.h>

#define NUM_HEADS   4
#define HEAD_DIM    64
#define HIDDEN      256
#define BLOCK_SZ    32
#define NUM_BLOCKS  64
#define B_BATCH     8
#define T_TOTAL     11008
#define SCALE_F     0.125f
#define INV_SCALE_F 8.0f
#define EPS_F       1e-6f

typedef __attribute__((ext_vector_type(2))) float v2f;
typedef __attribute__((ext_vector_type(8))) float v8f;

__device__ __forceinline__ float silu_f(float x) {
    return x / (1.0f + __expf(-x));
}

// ---------------------------------------------------------------------------
// Kernel 1: block-mean compressed K/V.  grid = B*NUM_BLOCKS, block = 256
// thread = (h*64 + d). Blocks past a sequence's length stay zero (ref pads).
// kcmp layout: [b][blk][h][d]   (dims contiguous -> v2f loads in score GEMM)
// vcmpT layout: [b][h][d][blk]  (blocks contiguous -> v2f loads in out GEMM)
// ---------------------------------------------------------------------------
__global__ __launch_bounds__(256)
void compress_kernel(const float* __restrict__ k, const float* __restrict__ v,
                     const int* __restrict__ offsets,
                     float* __restrict__ kcmp, float* __restrict__ vcmpT) {
    int bb  = blockIdx.x;
    int b   = bb >> 6;
    int blk = bb & 63;
    int h   = threadIdx.x >> 6;
    int d   = threadIdx.x & 63;
    int start = offsets[b];
    int len   = offsets[b + 1] - start;
    float ks = 0.0f, vs = 0.0f;
    if (blk * BLOCK_SZ < len) {
        const float* kp = k + (size_t)(start + blk * BLOCK_SZ) * HIDDEN + h * HEAD_DIM + d;
        const float* vp = v + (size_t)(start + blk * BLOCK_SZ) * HIDDEN + h * HEAD_DIM + d;
        #pragma unroll 8
        for (int j = 0; j < BLOCK_SZ; ++j) {
            ks += kp[(size_t)j * HIDDEN];
            vs += vp[(size_t)j * HIDDEN];
        }
        ks *= (1.0f / 32.0f);
        vs *= (1.0f / 32.0f);
    }
    kcmp[((size_t)(b * NUM_BLOCKS + blk) * NUM_HEADS + h) * HEAD_DIM + d] = ks;
    vcmpT[(((size_t)(b * NUM_HEADS + h)) * HEAD_DIM + d) * NUM_BLOCKS + blk] = vs;
}

// ---------------------------------------------------------------------------
// Kernel 2: g_cmp = sigmoid(q . Wg_cmp), one thread per (t,h)
// ---------------------------------------------------------------------------
__global__ __launch_bounds__(256)
void gate_kernel(const float* __restrict__ q, const float* __restrict__ wg,
                 float* __restrict__ g) {
    int i = blockIdx.x * 256 + threadIdx.x;
    if (i >= T_TOTAL * NUM_HEADS) return;
    int t = i >> 2, h = i & 3;
    const float* qp = q + (size_t)t * HIDDEN + h * HEAD_DIM;
    const float* wp = wg + h * HEAD_DIM;
    float s = 0.0f;
    #pragma unroll 16
    for (int d = 0; d < HEAD_DIM; ++d) s += qp[d] * wp[d];
    g[i] = 1.0f / (1.0f + __expf(-s));
}

// ---------------------------------------------------------------------------
// Compressed-attention WMMA bodies, specialized on the number of 16-block
// N-tiles (NT = 1..4) so every loop is static and fully unrolled: no guards
// around WMMAs, accumulators stay pinned, loads cluster ahead of the WMMAs.
//
// WMMA f32 16x16x4 operand mapping (CDNA5 ISA 7.12.2):
//   A[m][k]: lane = (k/2)*16 + m, vgpr = k%2
//   B[k][n]: lane = (k/2)*16 + n, vgpr = k%2
//   C/D[g][lane]: row = g + 8*(lane/16), col = lane%16
// ---------------------------------------------------------------------------
template <int NT>
__device__ __forceinline__ void cmp_scores(const float (*__restrict__ qsh)[65],
                                           const float* __restrict__ kc_base,
                                           int half, int idx, int pos0,
                                           float (*__restrict__ psh)[65]) {
    v8f acc[NT] = {};
    #pragma unroll
    for (int kk = 0; kk < 16; ++kk) {            // K over HEAD_DIM/4
        int kb = kk * 4 + 2 * half;
        v2f a;
        a[0] = qsh[idx][kb];
        a[1] = qsh[idx][kb + 1];
        #pragma unroll
        for (int nt = 0; nt < NT; ++nt) {
            int blk = nt * 16 + idx;
            v2f bb = *(const v2f*)(kc_base + (size_t)blk * HIDDEN + kb);
            acc[nt] = __builtin_amdgcn_wmma_f32_16x16x4_f32(
                false, a, false, bb, (short)0, acc[nt], false, false);
        }
    }
    #pragma unroll
    for (int nt = 0; nt < NT; ++nt) {
        int blk = nt * 16 + idx;
        #pragma unroll
        for (int g = 0; g < 8; ++g) {
            int m  = g + 8 * half;
            int qb = (pos0 + m) >> 5;
            float raw = acc[nt][g] * SCALE_F;
            psh[m][blk] = (blk <= qb) ? silu_f(raw) * INV_SCALE_F : 0.0f;
        }
    }
}

template <int NT>
__device__ __forceinline__ void cmp_output(const float (*__restrict__ psh)[65],
                                           const float* __restrict__ vcT_base,
                                           const float* __restrict__ gsh,
                                           float* __restrict__ ocmp,
                                           int t0, int h, int half, int idx) {
    v8f acc2[4] = {};
    #pragma unroll
    for (int kc = 0; kc < NT * 4; ++kc) {        // K over selected block range
        int kb = kc * 4 + 2 * half;
        v2f a;
        a[0] = psh[idx][kb];
        a[1] = psh[idx][kb + 1];
        #pragma unroll
        for (int dt = 0; dt < 4; ++dt) {
            int d = dt * 16 + idx;
            v2f bb = *(const v2f*)(vcT_base + (size_t)d * NUM_BLOCKS + kb);
            acc2[dt] = __builtin_amdgcn_wmma_f32_16x16x4_f32(
                false, a, false, bb, (short)0, acc2[dt], false, false);
        }
    }
    #pragma unroll
    for (int dt = 0; dt < 4; ++dt) {
        #pragma unroll
        for (int g = 0; g < 8; ++g) {
            int m = g + 8 * half;
            int d = dt * 16 + idx;
            ocmp[(size_t)(t0 + m) * HIDDEN + h * HEAD_DIM + d] = acc2[dt][g] * gsh[m];
        }
    }
}

// ---------------------------------------------------------------------------
// Kernel 3: compressed attention via fp32 WMMA + exact top-k.
// grid = T/16 tiles (tiles never cross batches: all offsets % 256 == 0),
// block = 128 (4 waves, wave h handles head h).
// ---------------------------------------------------------------------------
__global__ __launch_bounds__(128)
void cmp_attn_kernel(const float* __restrict__ q,
                     const float* __restrict__ kcmp,
                     const float* __restrict__ vcmpT,
                     const float* __restrict__ gcmp,
                     const int* __restrict__ batch_ids,
                     const int* __restrict__ pos_ids,
                     float* __restrict__ ocmp,
                     int* __restrict__ topsel) {
    __shared__ float qs[NUM_HEADS][16][65];   // padded stride: no 64-bank conflicts
    __shared__ float ps[NUM_HEADS][16][65];
    __shared__ float gs[NUM_HEADS][16];

    const int t0   = blockIdx.x * 16;
    const int h    = threadIdx.x >> 5;
    const int lane = threadIdx.x & 31;
    const int half = lane >> 4;
    const int idx  = lane & 15;

    const int b    = batch_ids[t0];
    const int pos0 = pos_ids[t0];

    // --- load q tile (16 rows x 64 dims) into LDS: 2 lanes per row ---
    {
        int row = lane >> 1;
        int seg = (lane & 1) * 32;
        const float* src = q + (size_t)(t0 + row) * HIDDEN + h * HEAD_DIM + seg;
        #pragma unroll
        for (int i = 0; i < 8; ++i) {
            float4 ld = *(const float4*)(src + i * 4);
            qs[h][row][seg + i * 4 + 0] = ld.x;
            qs[h][row][seg + i * 4 + 1] = ld.y;
            qs[h][row][seg + i * 4 + 2] = ld.z;
            qs[h][row][seg + i * 4 + 3] = ld.w;
        }
    }
    if (lane < 16) gs[h][lane] = gcmp[(t0 + lane) * NUM_HEADS + h];
    __syncthreads();

    const int qbmax  = (pos0 + 15) >> 5;      // highest causal block in tile
    const int ntiles = (qbmax >> 4) + 1;      // 1..4 tiles of 16 blocks

    const float* kc_base  = kcmp  + ((size_t)(b * NUM_BLOCKS) * NUM_HEADS + h) * HEAD_DIM;
    const float* vcT_base = vcmpT + ((size_t)(b * NUM_HEADS + h)) * HEAD_DIM * NUM_BLOCKS;

    switch (ntiles) {
        case 1: cmp_scores<1>(qs[h], kc_base, half, idx, pos0, ps[h]); break;
        case 2: cmp_scores<2>(qs[h], kc_base, half, idx, pos0, ps[h]); break;
        case 3: cmp_scores<3>(qs[h], kc_base, half, idx, pos0, ps[h]); break;
        default: cmp_scores<4>(qs[h], kc_base, half, idx, pos0, ps[h]); break;
    }
    __syncthreads();

    // --- exact top-k (k=4; ranks 1,3 are duplicates of 0,2 in the ref) ---
    // sel[j] = 1.0 at local block, p_cmp for j<qb, 0.0 for j>qb; all 64 entries
    // participate like jax.lax.top_k; ties -> lowest index via ascending scan.
    if (lane < 16) {
        int m  = lane;
        int qb = (pos0 + m) >> 5;
        unsigned long long taken = 0ull;
        int sel0 = 0, sel2 = 0;
        for (int pass = 0; pass < 3; ++pass) {
            float best = -3.4e38f;
            int bi = 0;
            for (int j = 0; j < NUM_BLOCKS; ++j) {
                if ((taken >> j) & 1ull) continue;
                float vv = (j > qb) ? 0.0f : ((j == qb) ? 1.0f : ps[h][m][j]);
                if (vv > best) { best = vv; bi = j; }
            }
            taken |= (1ull << bi);
            if (pass == 0) sel0 = bi;
            if (pass == 2) sel2 = bi;
        }
        int t = t0 + m;
        topsel[(t * NUM_HEADS + h) * 2 + 0] = sel0;
        topsel[(t * NUM_HEADS + h) * 2 + 1] = sel2;
    }

    switch (ntiles) {
        case 1: cmp_output<1>(ps[h], vcT_base, gs[h], ocmp, t0, h, half, idx); break;
        case 2: cmp_output<2>(ps[h], vcT_base, gs[h], ocmp, t0, h, half, idx); break;
        case 3: cmp_output<3>(ps[h], vcT_base, gs[h], ocmp, t0, h, half, idx); break;
        default: cmp_output<4>(ps[h], vcT_base, gs[h], ocmp, t0, h, half, idx); break;
    }
}

// ---------------------------------------------------------------------------
// Kernel 4: selected-block attention. grid = T, block = 128 (wave h = head h).
// Two distinct top-k blocks, each weighted 2x (ref duplicates ranks 1,3).
// ---------------------------------------------------------------------------
__global__ __launch_bounds__(128)
void slc_attn_kernel(const float* __restrict__ q,
                     const float* __restrict__ k,
                     const float* __restrict__ v,
                     const int* __restrict__ topsel,
                     const int* __restrict__ pos_ids,
                     float* __restrict__ oslc) {
    __shared__ float qs[NUM_HEADS][HEAD_DIM];
    const int t    = blockIdx.x;
    const int h    = threadIdx.x >> 5;
    const int lane = threadIdx.x & 31;
    const int pos  = pos_ids[t];
    const int seq_start = t - pos;            // tokens of a batch are contiguous

    if (lane < 16) {
        float4 qv = *(const float4*)(q + (size_t)t * HIDDEN + h * HEAD_DIM + lane * 4);
        qs[h][lane * 4 + 0] = qv.x;
        qs[h][lane * 4 + 1] = qv.y;
        qs[h][lane * 4 + 2] = qv.z;
        qs[h][lane * 4 + 3] = qv.w;
    }
    __syncthreads();

    const int b0 = topsel[(t * NUM_HEADS + h) * 2 + 0];
    const int b2 = topsel[(t * NUM_HEADS + h) * 2 + 1];

    float o0 = 0.0f, o1 = 0.0f;               // this lane owns dims lane, lane+32
    #pragma unroll
    for (int sblk = 0; sblk < 2; ++sblk) {
        int blk = sblk ? b2 : b0;
        int pin = blk * BLOCK_SZ + lane;      // lane = position within block
        int tgt = seq_start + pin;
        float p = 0.0f;
        if (pin <= pos) {                     // causal validity (also kills blk>qb)
            const float* kr = k + (size_t)tgt * HIDDEN + h * HEAD_DIM;
            float s = 0.0f;
            #pragma unroll
            for (int i = 0; i < 16; ++i) {
                float4 kv = *(const float4*)(kr + i * 4);
                s += kv.x * qs[h][i * 4 + 0] + kv.y * qs[h][i * 4 + 1]
                   + kv.z * qs[h][i * 4 + 2] + kv.w * qs[h][i * 4 + 3];
            }
            p = silu_f(s * SCALE_F) * INV_SCALE_F;
        }
        for (int j = 0; j < BLOCK_SZ; ++j) {
            float pj = __shfl(p, j, 32);      // uniform across wave after shfl
            int   tj = __shfl(tgt, j, 32);
            if (pj != 0.0f) {
                const float* vr = v + (size_t)tj * HIDDEN + h * HEAD_DIM;
                o0 += pj * vr[lane];
                o1 += pj * vr[lane + 32];
            }
        }
    }
    oslc[(size_t)t * HIDDEN + h * HEAD_DIM + lane]      = 2.0f * o0;
    oslc[(size_t)t * HIDDEN + h * HEAD_DIM + lane + 32] = 2.0f * o1;
}

// ---------------------------------------------------------------------------
// Kernel 5: two LayerNorms over HIDDEN=256 (cross-head), * u, sum.
// grid = T, block = 256.
// ---------------------------------------------------------------------------
__global__ __launch_bounds__(256)
void finalize_kernel(const float* __restrict__ ocmp, const float* __restrict__ oslc,
                     const float* __restrict__ u, float* __restrict__ out) {
    __shared__ float red[256];
    const int t = blockIdx.x;
    const int i = threadIdx.x;
    float xc = ocmp[(size_t)t * HIDDEN + i];
    float xs = oslc[(size_t)t * HIDDEN + i];

    auto blockSum = [&](float val) -> float {
        red[i] = val;
        __syncthreads();
        for (int off = 128; off > 0; off >>= 1) {
            if (i < off) red[i] += red[i + off];
            __syncthreads();
        }
        float r = red[0];
        __syncthreads();
        return r;
    };

    float muc = blockSum(xc) * (1.0f / HIDDEN);
    float vc  = blockSum((xc - muc) * (xc - muc)) * (1.0f / HIDDEN);
    float mus = blockSum(xs) * (1.0f / HIDDEN);
    float vs  = blockSum((xs - mus) * (xs - mus)) * (1.0f / HIDDEN);

    float uu = u[(size_t)t * HIDDEN + i];
    out[(size_t)t * HIDDEN + i] =
        ((xc - muc) * rsqrtf(vc + EPS_F) + (xs - mus) * rsqrtf(vs + EPS_F)) * uu;
}

// ---------------------------------------------------------------------------
extern "C" void kernel_launch(void* const* d_in, const int* in_sizes, int n_in,
                              void* d_out, int out_size, void* d_ws, size_t ws_size,
                              hipStream_t stream) {
    const float* q      = (const float*)d_in[0];
    const float* k      = (const float*)d_in[1];
    const float* v      = (const float*)d_in[2];
    const float* u      = (const float*)d_in[3];
    const float* wg_cmp = (const float*)d_in[4];
    // d_in[5] (Wg_slc) feeds a dead value in the reference; unused.
    const int* offs = (const int*)d_in[6];
    const int* bids = (const int*)d_in[7];
    const int* pids = (const int*)d_in[8];
    float* out = (float*)d_out;

    // workspace layout
    float* ws_f  = (float*)d_ws;
    float* ocmp  = ws_f;                                         // T*256
    float* oslc  = ocmp + (size_t)T_TOTAL * HIDDEN;              // T*256
    float* kcmp  = oslc + (size_t)T_TOTAL * HIDDEN;              // B*64*256
    float* vcmpT = kcmp + (size_t)B_BATCH * NUM_BLOCKS * HIDDEN; // B*4*64*64
    float* gcmp  = vcmpT + (size_t)B_BATCH * NUM_BLOCKS * HIDDEN; // T*4
    int*   tsel  = (int*)(gcmp + (size_t)T_TOTAL * NUM_HEADS);    // T*4*2

    compress_kernel<<<B_BATCH * NUM_BLOCKS, 256, 0, stream>>>(k, v, offs, kcmp, vcmpT);
    gate_kernel<<<(T_TOTAL * NUM_HEADS + 255) / 256, 256, 0, stream>>>(q, wg_cmp, gcmp);
    cmp_attn_kernel<<<T_TOTAL / 16, 128, 0, stream>>>(q, kcmp, vcmpT, gcmp, bids, pids,
                                                      ocmp, tsel);
    slc_attn_kernel<<<T_TOTAL, 128, 0, stream>>>(q, k, v, tsel, pids, oslc);
    finalize_kernel<<<T_TOTAL, 256, 0, stream>>>(ocmp, oslc, u, out);
}